// TripletLoss_47742856462444
// MI455X (gfx1250) — compile-verified
//
#include <hip/hip_runtime.h>
#include <math.h>

typedef __attribute__((ext_vector_type(2))) float v2f;
typedef __attribute__((ext_vector_type(8))) float v8f;

#define N 320
#define D 256
#define MIN_NORM 1e-15f

// Workspace layout (float units):
//   bytes [0..7]   : double  sum accumulator
//   float [2]      : uint    count accumulator
//   float [16..)   : x2[N]      = ||offset_i||^2
//   float [16+N..) : pn[N]      = <offset_j, normal_j>
//   float [16+2N..): an[N]      = ||normal_j||
//   float [1024..) : dm[N*N]
#define WS_X2  16
#define WS_PN  (16 + N)
#define WS_AN  (16 + 2 * N)
#define WS_DM  1024

// ---------------------------------------------------------------------------
// Kernel 1: per-row stats + accumulator init. One block (D threads) per row.
// ---------------------------------------------------------------------------
__global__ void tl_rowstats(const float* __restrict__ off,
                            const float* __restrict__ nrm,
                            float* __restrict__ ws) {
    __shared__ float s0[D], s1[D], s2[D];
    const int row = blockIdx.x;
    const int t   = threadIdx.x;

    if (row == 0 && t == 0) {
        *(double*)ws = 0.0;                    // sum
        ((unsigned int*)ws)[2] = 0u;           // count
    }

    const float o = off[row * D + t];
    const float n = nrm[row * D + t];
    s0[t] = o * o;
    s1[t] = o * n;
    s2[t] = n * n;
    __syncthreads();
    for (int s = D / 2; s > 0; s >>= 1) {
        if (t < s) {
            s0[t] += s0[t + s];
            s1[t] += s1[t + s];
            s2[t] += s2[t + s];
        }
        __syncthreads();
    }
    if (t == 0) {
        ws[WS_X2 + row] = s0[0];
        ws[WS_PN + row] = s1[0];
        ws[WS_AN + row] = sqrtf(s2[0]);
    }
}

// ---------------------------------------------------------------------------
// Kernel 2: dm[i,j] via dual fp32 WMMA GEMMs (gram = X·Xt, xn = X·Nt) with a
// fused hyperbolic-distance epilogue. One wave32 per 16x16 output tile.
// ---------------------------------------------------------------------------
__global__ __launch_bounds__(32)
void tl_dm_wmma(const float* __restrict__ off,
                const float* __restrict__ nrm,
                const float* __restrict__ curv,
                float* __restrict__ ws) {
    const float* x2 = ws + WS_X2;
    const float* pn = ws + WS_PN;
    const float* an = ws + WS_AN;
    float*       dm = ws + WS_DM;

    const int bi   = blockIdx.x;          // tile row
    const int bj   = blockIdx.y;          // tile col
    const int lane = threadIdx.x;         // 0..31, wave32
    const int half = lane >> 4;           // 0: K=0,1  1: K=2,3
    const int l    = lane & 15;

    // A: rows of offset for tile bi; B: rows of offset/normal for tile bj
    const float* arow = off + (size_t)(bi * 16 + l) * D + half * 2;
    const float* borw = off + (size_t)(bj * 16 + l) * D + half * 2;
    const float* bnrw = nrm + (size_t)(bj * 16 + l) * D + half * 2;

    v8f cg = {};   // gram accumulator
    v8f cn = {};   // x·normal^T accumulator
    for (int k = 0; k < D; k += 4) {
        v2f a  = *(const v2f*)(arow + k);
        v2f bo = *(const v2f*)(borw + k);
        v2f bn = *(const v2f*)(bnrw + k);
        cg = __builtin_amdgcn_wmma_f32_16x16x4_f32(false, a, false, bo,
                                                   (short)0, cg, false, false);
        cn = __builtin_amdgcn_wmma_f32_16x16x4_f32(false, a, false, bn,
                                                   (short)0, cn, false, false);
    }

    // Epilogue. C/D layout: lane l (col j = bj*16 + l%16), VGPR r -> row
    // m = r + (lane>=16 ? 8 : 0).
    const float c      = curv[0];
    const float sqrt_c = sqrtf(c);
    const int   j      = bj * 16 + l;
    const float x2j    = x2[j];
    const float confj  = 1.0f - c * x2j;       // conformal factor (== coef_v)
    const float pnj    = pn[j];
    const float anj    = fmaxf(fabsf(confj) * an[j], MIN_NORM);

#pragma unroll
    for (int r = 0; r < 8; ++r) {
        const int   i    = bi * 16 + r + half * 8;
        const float g    = cg[r];               // <x_i, p_j>
        const float xnij = cn[r];               // <x_i, n_j>
        const float x2i  = x2[i];
        const float uv   = -g;                  // <u=-p_j, v=x_i>
        const float cu   = 1.0f + 2.0f * c * uv + c * x2i;
        const float cv   = confj;               // 1 - c*x2[j]
        float denom = 1.0f + 2.0f * c * uv + c * c * x2i * x2j;
        denom = fmaxf(denom, MIN_NORM);
        float dn2 = (cu * cu * x2j - 2.0f * cu * cv * g + cv * cv * x2i)
                    / (denom * denom);
        dn2 = fmaxf(dn2, MIN_NORM);
        const float sc  = confj * (cv * xnij - cu * pnj) / denom;
        const float num = 2.0f * sqrt_c * sc;
        const float den = fmaxf((1.0f - c * dn2) * anj, MIN_NORM);
        const float dist = asinhf(num / den) / sqrt_c;
        dm[i * N + j] = expf(dist);
    }
}

// ---------------------------------------------------------------------------
// Kernel 3: masked triplet reduction over N^3. Each thread handles 16
// consecutive k for one (i,j): 16 mask bytes (uint4) + 64B of L2-resident dm.
// ---------------------------------------------------------------------------
__global__ __launch_bounds__(256)
void tl_reduce(const unsigned char* __restrict__ mask,
               float* __restrict__ ws) {
    const float* dm = ws + WS_DM;
    const int KC    = N / 16;                 // 20 chunks per (i,j)
    const long long total = (long long)N * N * KC;   // 2,048,000

    float        lsum = 0.0f;
    unsigned int lcnt = 0u;

    for (long long t = (long long)blockIdx.x * blockDim.x + threadIdx.x;
         t < total;
         t += (long long)gridDim.x * blockDim.x) {
        const int       kc = (int)(t % KC);
        const long long ij = t / KC;
        const int       j  = (int)(ij % N);
        const int       i  = (int)(ij / N);
        const int       k0 = kc * 16;

        const float dij = dm[i * N + j];
        const uint4 m = *(const uint4*)(mask + ((long long)i * N + j) * N + k0);
        const unsigned int mw[4] = {m.x, m.y, m.z, m.w};
        const float* dr = dm + i * N + k0;
#pragma unroll
        for (int w = 0; w < 4; ++w) {
#pragma unroll
            for (int b = 0; b < 4; ++b) {
                if ((mw[w] >> (8 * b)) & 0xFFu) {
                    const float tl = dr[w * 4 + b] - dij;   // margin = 0
                    if (tl > 1e-16f) { lsum += tl; lcnt++; }
                }
            }
        }
    }

    __shared__ float        ssum[256];
    __shared__ unsigned int scnt[256];
    const int t = threadIdx.x;
    ssum[t] = lsum;
    scnt[t] = lcnt;
    __syncthreads();
    for (int s = 128; s > 0; s >>= 1) {
        if (t < s) {
            ssum[t] += ssum[t + s];
            scnt[t] += scnt[t + s];
        }
        __syncthreads();
    }
    if (t == 0) {
        atomicAdd((double*)ws, (double)ssum[0]);
        atomicAdd(((unsigned int*)ws) + 2, scnt[0]);
    }
}

// ---------------------------------------------------------------------------
// Kernel 4: finalize loss = sum / max(count, 1)
// ---------------------------------------------------------------------------
__global__ void tl_finalize(const float* __restrict__ ws,
                            float* __restrict__ out) {
    const double       s = *(const double*)ws;
    const unsigned int c = ((const unsigned int*)ws)[2];
    out[0] = (float)(s / (double)(c > 0u ? c : 1u));
}

extern "C" void kernel_launch(void* const* d_in, const int* in_sizes, int n_in,
                              void* d_out, int out_size, void* d_ws, size_t ws_size,
                              hipStream_t stream) {
    const float*         off  = (const float*)d_in[0];
    const float*         nrm  = (const float*)d_in[1];
    const float*         curv = (const float*)d_in[2];
    const unsigned char* mask = (const unsigned char*)d_in[3];
    float*               out  = (float*)d_out;
    float*               ws   = (float*)d_ws;

    tl_rowstats<<<N, D, 0, stream>>>(off, nrm, ws);

    dim3 g2(N / 16, N / 16);
    tl_dm_wmma<<<g2, 32, 0, stream>>>(off, nrm, curv, ws);

    tl_reduce<<<8000, 256, 0, stream>>>(mask, ws);

    tl_finalize<<<1, 1, 0, stream>>>(ws, out);
}